// FFM_73847667687628
// MI455X (gfx1250) — compile-verified
//
#include <hip/hip_runtime.h>
#include <hip/hip_bf16.h>
#include <stdint.h>

#define FIELDS 39
#define EMB 16
#define ROWF (FIELDS * EMB)                 // 624 floats per gathered block
#define ROWB (ROWF * 4)                     // 2496 bytes per gathered block
#define CHUNKS_PER_BLK (ROWB / 16)          // 156 x 16B chunks per block
#define NCHUNK (FIELDS * CHUNKS_PER_BLK)    // 6084 chunks per batch element
#define NTILES 3                            // ceil(39/16)
#define NTASKS (FIELDS * NTILES)            // 117 (j, i-tile) WMMA tasks
#define PBLKS 48                            // padded block count (3 full tiles)

typedef float v2f __attribute__((ext_vector_type(2)));
typedef float v8f __attribute__((ext_vector_type(8)));

// Low 32 bits of the flat address of an LDS object == byte offset within the
// workgroup's LDS allocation (flat->LDS aperture truncation, ISA 10.2).
__device__ __forceinline__ uint32_t lds_off32(const void* p) {
  return (uint32_t)(uintptr_t)p;
}

extern "C" __global__ __launch_bounds__(256)
void ffm_kernel(const int* __restrict__ fidx,
                const float* __restrict__ fval,
                const float* __restrict__ femb,
                const float* __restrict__ w1,
                const float* __restrict__ bias,
                float* __restrict__ out)
{
  extern __shared__ float smem[];
  float* sE  = smem;                         // PBLKS*624 floats: E blocks (39 live + 9 zero pad)
  float* sV  = sE + PBLKS * ROWF;            // 48: feature values, zero padded
  float* sFO = sV + PBLKS;                   // 39: first-order contributions
  int*   sIdx = (int*)(sFO + FIELDS);        // 39: feature indices
  float* sRed = (float*)(sIdx + FIELDS);     // 256: block reduction

  const int b   = blockIdx.x;
  const int tid = threadIdx.x;

  if (tid < FIELDS) {
    int   id = fidx[b * FIELDS + tid];
    float v  = fval[b * FIELDS + tid];
    sIdx[tid] = id;
    sV[tid]   = v;
    sFO[tid]  = v * w1[id];                  // first-order gather
  } else if (tid < PBLKS) {
    sV[tid] = 0.0f;                          // zero pad: kills i >= 39 terms
  }
  // Zero the pad blocks (rows 39..47 of the i dimension read zeros later).
  for (int z = tid; z < (PBLKS - FIELDS) * ROWF; z += 256)
    sE[FIELDS * ROWF + z] = 0.0f;
  __syncthreads();

  // ---- Async DMA gather: 39 blocks x 2496B -> LDS (16B per lane per issue) ----
  {
    const uint32_t sE_base = lds_off32(sE);
    const uint64_t gbase   = (uint64_t)(uintptr_t)femb;
    for (int c = tid; c < NCHUNK; c += 256) {
      const int i = c / CHUNKS_PER_BLK;
      const int r = c - i * CHUNKS_PER_BLK;
      const uint32_t go = (uint32_t)sIdx[i] * (uint32_t)ROWB + (uint32_t)r * 16u;
      const uint32_t lo = sE_base + (uint32_t)i * (uint32_t)ROWB + (uint32_t)r * 16u;
      asm volatile("global_load_async_to_lds_b128 %0, %1, %2"
                   :: "v"(lo), "v"(go), "s"(gbase) : "memory");
    }
    asm volatile("s_wait_asynccnt 0" ::: "memory");
  }
  __syncthreads();

  float acc = 0.0f;
  const int wave = tid >> 5;
  const int lane = tid & 31;
  const int row  = lane & 15;
  const int hi   = lane >> 4;

#if __has_builtin(__builtin_amdgcn_wmma_f32_16x16x4_f32)
  // D = A * B^T per (j, i-tile): A[m,k]=E_{i_m}[j,k], B[k,n]=E_j[i_n,k].
  // diag(D)[i] = S[i,j]. Sum over all ordered i!=j, then halve (S symmetric).
  const bool dvalid = (lane < 8) || (lane >= 24);   // lanes that hold a diagonal elem
  const int  m      = (lane < 8) ? lane : (lane - 16);
  const int  e      = lane & 7;                     // element index within C for diag

  for (int t = wave; t < NTASKS; t += 8) {
    const int j    = t / NTILES;
    const int tile = t - j * NTILES;
    const int ii   = tile * 16 + row;                // 0..47, pad reads zeros
    const float* arow = &sE[ii * ROWF + j * EMB];    // E_ii[j, 0..15]
    const float* brow = &sE[j * ROWF + ii * EMB];    // E_j[ii, 0..15]

    v8f c = {};
#pragma unroll
    for (int s = 0; s < 4; ++s) {
      // A 16x4 f32 layout: VGPR0 K = 2*hi, VGPR1 K = 2*hi+1 (per K-step s)
      const v2f a = *(const v2f*)(arow + 4 * s + 2 * hi);   // ds_load_b64
      // B 4x16 f32 layout: VGPR0 row K = hi, VGPR1 row K = hi+2
      v2f bb;
      bb.x = brow[4 * s + hi];
      bb.y = brow[4 * s + hi + 2];
      c = __builtin_amdgcn_wmma_f32_16x16x4_f32(
          /*neg_a=*/false, a, /*neg_b=*/false, bb,
          /*c_mod=*/(short)0, c, /*reuse_a=*/false, /*reuse_b=*/false);
    }

    // Branch-free diagonal extract: binary cndmask tree on bits of e.
    const float x0 = (e & 4) ? c[4] : c[0];
    const float x1 = (e & 4) ? c[5] : c[1];
    const float x2 = (e & 4) ? c[6] : c[2];
    const float x3 = (e & 4) ? c[7] : c[3];
    const float y0 = (e & 2) ? x2 : x0;
    const float y1 = (e & 2) ? x3 : x1;
    const float d  = (e & 1) ? y1 : y0;

    // Unconditional weights: sV pad (i >= 39 -> 0) removes the range check;
    // remaining condition applied as a single value select.
    const int   ig  = tile * 16 + m;                 // 0..47 (pad weight = 0)
    const float wgt = sV[ig] * sV[j];
    acc += (dvalid && (ig != j)) ? (d * wgt) : 0.0f;
  }
  acc *= 0.5f;   // ordered (i,j) i!=j counted twice
#else
  // Scalar fallback: one upper-triangle pair per thread iteration.
  for (int p = tid; p < (FIELDS * (FIELDS - 1)) / 2; p += 256) {
    int i = 0, rem = p;
    while (rem >= FIELDS - 1 - i) { rem -= FIELDS - 1 - i; ++i; }
    const int j = i + 1 + rem;
    const float* u = &sE[i * ROWF + j * EMB];
    const float* w = &sE[j * ROWF + i * EMB];
    float s = 0.0f;
#pragma unroll
    for (int ee = 0; ee < EMB; ++ee) s += u[ee]*w[ee];
    acc += s * sV[i] * sV[j];
  }
#endif

  // ---- Block reduction + epilogue ----
  sRed[tid] = acc;
  __syncthreads();
  for (int s2 = 128; s2 > 0; s2 >>= 1) {
    if (tid < s2) sRed[tid] += sRed[tid + s2];
    __syncthreads();
  }
  if (tid == 0) {
    float fo = 0.0f;
#pragma unroll
    for (int i = 0; i < FIELDS; ++i) fo += sFO[i];
    out[b] = sRed[0] + fo + bias[0];
  }
}

extern "C" void kernel_launch(void* const* d_in, const int* in_sizes, int n_in,
                              void* d_out, int out_size, void* d_ws, size_t ws_size,
                              hipStream_t stream) {
  (void)n_in; (void)d_ws; (void)ws_size; (void)out_size;
  const int*   fidx = (const int*)d_in[0];
  const float* fval = (const float*)d_in[1];
  const float* femb = (const float*)d_in[2];
  const float* w1   = (const float*)d_in[3];
  const float* bias = (const float*)d_in[4];
  float* out = (float*)d_out;

  const int batch = in_sizes[0] / FIELDS;   // 4096
  const size_t shmem =
      (size_t)(PBLKS * ROWF + PBLKS + 2 * FIELDS + 256) * sizeof(float);

  ffm_kernel<<<batch, 256, shmem, stream>>>(fidx, fval, femb, w1, bias, out);
}